// FocusedLinearAttention_16492674416937
// MI455X (gfx1250) — compile-verified
//
#include <hip/hip_runtime.h>
#include <math.h>
#include <stdint.h>

typedef float v2f __attribute__((ext_vector_type(2)));
typedef float v8f __attribute__((ext_vector_type(8)));

#define EPSV 1e-06f
#define Bn 4
#define Nn 1024
#define Cn 512
#define Kn 10
#define Tn 11
#define Hn 8
#define Dn 64
#define Mn 4096
#define KCONV (Tn * Cn) /* 5632 */

#define BM 128
#define BN 64
#define BK 32
#define BKP (BK + 4)   /* padded A row: 36 floats, 144B (16B aligned, bank-skewed) */
#define BNP (BN + 4)   /* padded B row: 68 floats, 272B */

// --------------------------------------------------------------------------
// CDNA5 async global->LDS path (ASYNCcnt-tracked), with safe fallback.
// Builtin signature (from hipcc diagnostic): (v4i AS1*, v4i AS3*, Ii, Ii).
// --------------------------------------------------------------------------
#if defined(__has_builtin)
#if __has_builtin(__builtin_amdgcn_global_load_async_to_lds_b128)
#define USE_ASYNC_LDS 1
#endif
#endif
#ifndef USE_ASYNC_LDS
#define USE_ASYNC_LDS 0
#endif

typedef int ai4 __attribute__((vector_size(16)));
typedef __attribute__((address_space(1))) ai4 gi4;
typedef __attribute__((address_space(3))) ai4 li4;

__device__ __forceinline__ void async_wait_all()
{
#if USE_ASYNC_LDS
#if __has_builtin(__builtin_amdgcn_s_wait_asynccnt)
    __builtin_amdgcn_s_wait_asynccnt(0);
#else
    asm volatile("s_wait_asynccnt 0x0" ::: "memory");
#endif
#endif
}

// Copy 16 bytes global -> LDS (async DMA path when available).
__device__ __forceinline__ void copy16(const float* __restrict__ g, float* l)
{
#if USE_ASYNC_LDS
    __builtin_amdgcn_global_load_async_to_lds_b128(
        (gi4*)(uintptr_t)g,
        (li4*)(uint32_t)(uintptr_t)l,
        0, 0);
#else
    *(float4*)l = *(const float4*)g;
#endif
}

// ---------------------------------------------------------------------------
// GEMM with bias: Cout[M,N] = A[M,Kd] @ Bw[Kd,N] + bias[N]
// GATHER=true: A rows are gathered per tap from v via adjacency
//   A[m, t*C+c] = (t==0) ? v[b,n,c] : v[b, adj[b,n,t-1], c], Kd = 11*512.
// 256 threads = 8 waves; block tile 128x64; wave tile 32x32 = 2x2 WMMA f32;
// double-buffered LDS with async global->LDS prefetch of the next K-chunk.
// ---------------------------------------------------------------------------
template <bool GATHER>
__global__ __launch_bounds__(256) void gemm_wmma(
    const float* __restrict__ A, const int* __restrict__ adj,
    const float* __restrict__ Bw, const float* __restrict__ bias,
    float* __restrict__ Cout, int M, int N, int Kd)
{
    __shared__ float As[2][BM][BKP];
    __shared__ float Bs[2][BK][BNP];

    const int tid  = threadIdx.x;
    const int wave = tid >> 5;
    const int lane = tid & 31;
    const int half = lane >> 4;
    const int lm   = lane & 15;
    const int wm   = wave >> 1;   // 0..3 -> M
    const int wn   = wave & 1;    // 0..1 -> N
    const int m0   = blockIdx.y * BM;
    const int n0   = blockIdx.x * BN;

    v8f acc[2][2];
    #pragma unroll
    for (int i = 0; i < 2; ++i)
        #pragma unroll
        for (int j = 0; j < 2; ++j)
            #pragma unroll
            for (int r = 0; r < 8; ++r) acc[i][j][r] = 0.0f;

    auto load_tiles = [&](int buf, int kc) {
        // A tile: 128 x 32 floats = 1024 float4 chunks
        #pragma unroll
        for (int i = 0; i < 4; ++i) {
            int e = tid + i * 256;
            int r = e >> 3, cch = e & 7;
            const float* src;
            if (GATHER) {
                int t = kc >> 9, cc0 = kc & 511;
                int mrow = m0 + r;
                int b = mrow >> 10, n = mrow & 1023;
                int sr = (t == 0) ? n : adj[((b << 10) + n) * Kn + (t - 1)];
                src = A + ((size_t)(b << 10) + sr) * Cn + cc0 + cch * 4;
            } else {
                src = A + (size_t)(m0 + r) * Kd + kc + cch * 4;
            }
            copy16(src, &As[buf][r][cch * 4]);
        }
        // B tile: 32 x 64 floats = 512 float4 chunks
        #pragma unroll
        for (int i = 0; i < 2; ++i) {
            int e = tid + i * 256;
            int r = e >> 4, cch = e & 15;
            copy16(Bw + (size_t)(kc + r) * N + n0 + cch * 4, &Bs[buf][r][cch * 4]);
        }
    };

    const int nk = Kd / BK;
    load_tiles(0, 0);
    async_wait_all();
    __syncthreads();

    for (int ik = 0; ik < nk; ++ik) {
        const int cur = ik & 1;
        if (ik + 1 < nk) load_tiles(cur ^ 1, (ik + 1) * BK);

        #pragma unroll
        for (int kk = 0; kk < BK; kk += 4) {
            int ks = kk + 2 * half;
            v2f af[2], bf[2];
            #pragma unroll
            for (int tm = 0; tm < 2; ++tm) {
                int row = wm * 32 + tm * 16 + lm;
                af[tm].x = As[cur][row][ks];
                af[tm].y = As[cur][row][ks + 1];
            }
            #pragma unroll
            for (int tn = 0; tn < 2; ++tn) {
                int col = wn * 32 + tn * 16 + lm;
                bf[tn].x = Bs[cur][ks][col];
                bf[tn].y = Bs[cur][ks + 1][col];
            }
            #pragma unroll
            for (int tm = 0; tm < 2; ++tm)
                #pragma unroll
                for (int tn = 0; tn < 2; ++tn)
                    acc[tm][tn] = __builtin_amdgcn_wmma_f32_16x16x4_f32(
                        false, af[tm], false, bf[tn], (short)0, acc[tm][tn],
                        false, false);
        }
        async_wait_all();
        __syncthreads();
    }

    #pragma unroll
    for (int tm = 0; tm < 2; ++tm)
        #pragma unroll
        for (int tn = 0; tn < 2; ++tn) {
            int colg = n0 + wn * 32 + tn * 16 + lm;
            float bvv = bias ? bias[colg] : 0.0f;
            #pragma unroll
            for (int r = 0; r < 8; ++r) {
                int rowg = m0 + wm * 32 + tm * 16 + half * 8 + r;
                Cout[(size_t)rowg * N + colg] = acc[tm][tn][r] + bvv;
            }
        }
}

// ---------------------------------------------------------------------------
// Wt[(t*C + c)*C + o] = Wconv[o*(C*T) + c*T + t]
// ---------------------------------------------------------------------------
__global__ void repack_wconv(const float* __restrict__ Wconv, float* __restrict__ Wt)
{
    size_t idx = (size_t)blockIdx.x * 256 + threadIdx.x;
    if (idx >= (size_t)Cn * Cn * Tn) return;
    int o = (int)(idx & 511);
    int rest = (int)(idx >> 9);   // t*C + c
    int c = rest & 511;
    int t = rest >> 9;
    Wt[idx] = Wconv[(size_t)o * KCONV + (size_t)c * Tn + t];
}

// ---------------------------------------------------------------------------
// relu+eps, divide by softplus(scale), "focus": x <- (x/(||x||+eps))^3 * ||x||
// ---------------------------------------------------------------------------
__global__ __launch_bounds__(256) void focus_kernel(float* __restrict__ x,
                                                    const float* __restrict__ scale_p)
{
    __shared__ float red[256];
    const int m = blockIdx.x;
    const int tid = threadIdx.x;
    float* row = x + (size_t)m * Cn;

    float a0 = row[tid];
    float a1 = row[tid + 256];
    float s0 = log1pf(expf(scale_p[tid]));
    float s1 = log1pf(expf(scale_p[tid + 256]));
    a0 = (a0 > 0.0f ? a0 : 0.0f) + EPSV;
    a1 = (a1 > 0.0f ? a1 : 0.0f) + EPSV;
    a0 /= s0;
    a1 /= s1;

    red[tid] = a0 * a0 + a1 * a1;
    __syncthreads();
    for (int s = 128; s > 0; s >>= 1) {
        if (tid < s) red[tid] += red[tid + s];
        __syncthreads();
    }
    float nrm = sqrtf(red[0]);
    float inv = 1.0f / (nrm + EPSV);
    float u0 = a0 * inv, u1 = a1 * inv;
    row[tid]       = u0 * u0 * u0 * nrm;
    row[tid + 256] = u1 * u1 * u1 * nrm;
}

// kmean[b*C + c] = mean over n of k[b,n,c]
__global__ void kmean_kernel(const float* __restrict__ k, float* __restrict__ kmean)
{
    int idx = blockIdx.x * 256 + threadIdx.x;   // 0..B*C-1
    int b = idx >> 9;
    int c = idx & 511;
    const float* p = k + (size_t)b * Nn * Cn + c;
    float s = 0.0f;
    for (int n = 0; n < Nn; ++n) s += p[(size_t)n * Cn];
    kmean[idx] = s * (1.0f / (float)Nn);
}

// kv[bh, d, e] = (1/N) sum_n k[b,n,h*64+d] * v[b,n,h*64+e]
__global__ __launch_bounds__(256) void kv_kernel(const float* __restrict__ k,
                                                 const float* __restrict__ v,
                                                 float* __restrict__ kv)
{
    __shared__ float ks[8][64];
    __shared__ float vs[8][64];
    const int bh = blockIdx.x;
    const int b = bh >> 3, h = bh & 7;
    const int tid = threadIdx.x;
    const int e  = tid & 63;
    const int d0 = (tid >> 6) * 16;
    const float* kb = k + (size_t)b * Nn * Cn + h * Dn;
    const float* vb = v + (size_t)b * Nn * Cn + h * Dn;

    float acc[16];
    #pragma unroll
    for (int j = 0; j < 16; ++j) acc[j] = 0.0f;

    for (int n0 = 0; n0 < Nn; n0 += 8) {
        #pragma unroll
        for (int i = 0; i < 2; ++i) {
            int idx = tid + i * 256;
            int r = idx >> 6, c = idx & 63;
            ks[r][c] = kb[(size_t)(n0 + r) * Cn + c];
            vs[r][c] = vb[(size_t)(n0 + r) * Cn + c];
        }
        __syncthreads();
        #pragma unroll
        for (int r = 0; r < 8; ++r) {
            float ve = vs[r][e];
            #pragma unroll
            for (int j = 0; j < 16; ++j) acc[j] += ks[r][d0 + j] * ve;
        }
        __syncthreads();
    }
    #pragma unroll
    for (int j = 0; j < 16; ++j)
        kv[(size_t)bh * (Dn * Dn) + (d0 + j) * Dn + e] = acc[j] * (1.0f / (float)Nn);
}

// x[m,c] = (q_h . kv_h[:,e]) * z_h + x_local[m,c]
__global__ __launch_bounds__(256) void attn_kernel(const float* __restrict__ q,
                                                   const float* __restrict__ kv,
                                                   const float* __restrict__ kmean,
                                                   const float* __restrict__ xl,
                                                   float* __restrict__ x)
{
    __shared__ float qs[Cn];
    __shared__ float zs[Hn];
    const int m = blockIdx.x;
    const int b = m >> 10;
    const int tid = threadIdx.x;
    qs[tid]       = q[(size_t)m * Cn + tid];
    qs[tid + 256] = q[(size_t)m * Cn + tid + 256];
    __syncthreads();

    const int g = tid >> 5, lane = tid & 31;   // wave g handles head g
    float p = qs[g * 64 + lane]      * kmean[b * Cn + g * 64 + lane]
            + qs[g * 64 + lane + 32] * kmean[b * Cn + g * 64 + lane + 32];
    for (int off = 16; off > 0; off >>= 1) p += __shfl_xor(p, off, 32);
    if (lane == 0) zs[g] = 1.0f / (p + EPSV);
    __syncthreads();

    #pragma unroll
    for (int i = 0; i < 2; ++i) {
        int c = tid + i * 256;
        int h = c >> 6, e = c & 63;
        const float* kvp = kv + ((size_t)(b * Hn + h)) * (Dn * Dn) + e;
        float a = 0.0f;
        #pragma unroll 8
        for (int d = 0; d < Dn; ++d) a += qs[h * 64 + d] * kvp[(size_t)d * Dn];
        x[(size_t)m * Cn + c] = a * zs[h] + xl[(size_t)m * Cn + c];
    }
}

// Disagreement closed form per (b,h): ||sum_n hn||^2 - sum_n ||hn||^2.
__global__ void disag_kernel(const float* __restrict__ src,
                             long long per_b, long long per_h,
                             int vstride, int nvec, float* __restrict__ part)
{
    const int blk = blockIdx.x;
    const int b = blk >> 3, h = blk & 7;
    const float* p = src + (size_t)b * per_b + (size_t)h * per_h;
    const int l = threadIdx.x & 31;
    float s0 = 0.0f, s1 = 0.0f, dg = 0.0f;
    for (int n = 0; n < nvec; ++n) {
        float a0 = p[(size_t)n * vstride + l];
        float a1 = p[(size_t)n * vstride + l + 32];
        float sq = a0 * a0 + a1 * a1;
        for (int off = 16; off > 0; off >>= 1) sq += __shfl_xor(sq, off, 32);
        float nrm = sqrtf(sq);
        float inv = 1.0f / fmaxf(nrm, 1e-12f);
        float h0 = a0 * inv, h1 = a1 * inv;
        s0 += h0; s1 += h1;
        dg += h0 * h0 + h1 * h1;
    }
    float val = (s0 * s0 + s1 * s1) - dg;
    for (int off = 16; off > 0; off >>= 1) val += __shfl_xor(val, off, 32);
    if (l == 0) part[blk] = val;
}

__global__ void loss_final(const float* __restrict__ part, float* __restrict__ out_loss)
{
    if (threadIdx.x == 0) {
        float L1 = 0.0f, L2 = 0.0f, L3 = 0.0f;
        for (int i = 0; i < 32; ++i)  L1 += part[i];        // local (v heads)
        for (int i = 32; i < 64; ++i) L2 += part[i];        // kv
        for (int i = 64; i < 96; ++i) L3 += part[i];        // out
        const float nBig = (float)Bn * Hn * (float)Nn * (float)Nn;
        const float nKv  = (float)Bn * Hn * (float)Dn * (float)Dn;
        *out_loss = (L1 / nBig + L3 / nBig + L2 / nKv) / 3.0f;
    }
}

// LayerNorm over C per row.
__global__ __launch_bounds__(256) void ln_kernel(const float* __restrict__ x,
                                                 const float* __restrict__ g,
                                                 const float* __restrict__ bln,
                                                 float* __restrict__ xn)
{
    __shared__ float red[256];
    const int m = blockIdx.x;
    const int tid = threadIdx.x;
    const float* row = x + (size_t)m * Cn;
    float v0 = row[tid], v1 = row[tid + 256];

    red[tid] = v0 + v1;
    __syncthreads();
    for (int s = 128; s > 0; s >>= 1) {
        if (tid < s) red[tid] += red[tid + s];
        __syncthreads();
    }
    float mu = red[0] * (1.0f / (float)Cn);
    __syncthreads();

    float d0 = v0 - mu, d1 = v1 - mu;
    red[tid] = d0 * d0 + d1 * d1;
    __syncthreads();
    for (int s = 128; s > 0; s >>= 1) {
        if (tid < s) red[tid] += red[tid + s];
        __syncthreads();
    }
    float inv = rsqrtf(red[0] * (1.0f / (float)Cn) + 1e-05f);
    xn[(size_t)m * Cn + tid]       = d0 * inv * g[tid] + bln[tid];
    xn[(size_t)m * Cn + tid + 256] = d1 * inv * g[tid + 256] + bln[tid + 256];
}

// ---------------------------------------------------------------------------
extern "C" void kernel_launch(void* const* d_in, const int* in_sizes, int n_in,
                              void* d_out, int out_size, void* d_ws, size_t ws_size,
                              hipStream_t stream)
{
    const float* Q       = (const float*)d_in[0];
    const int*   adj     = (const int*)  d_in[1];
    const float* Wq      = (const float*)d_in[2];
    const float* bq      = (const float*)d_in[3];
    const float* Wk      = (const float*)d_in[4];
    const float* bk      = (const float*)d_in[5];
    const float* Wv      = (const float*)d_in[6];
    const float* bv      = (const float*)d_in[7];
    const float* scale_p = (const float*)d_in[8];
    const float* Wconv   = (const float*)d_in[9];
    const float* bconv   = (const float*)d_in[10];
    const float* W2      = (const float*)d_in[11];
    const float* b2      = (const float*)d_in[12];
    const float* ln_g    = (const float*)d_in[13];
    const float* ln_b    = (const float*)d_in[14];
    const float* Wo      = (const float*)d_in[15];
    const float* bo      = (const float*)d_in[16];
    float* out = (float*)d_out;

    const size_t MC = (size_t)Mn * Cn;          // 2,097,152
    float* ws    = (float*)d_ws;
    float* qb    = ws;
    float* kb    = ws + MC;
    float* vb    = ws + 2 * MC;
    float* convb = ws + 3 * MC;                 // conv output, later reused as x
    float* xlb   = ws + 4 * MC;                 // x_local
    float* wtb   = ws + 5 * MC;                 // transposed Wconv, later xn
    float* kvb   = wtb + (size_t)KCONV * Cn;    // B*H*64*64
    float* kmb   = kvb + (size_t)Bn * Hn * Dn * Dn;
    float* partb = kmb + (size_t)Bn * Cn;       // 96 partials

    dim3 gdim(Cn / BN, Mn / BM);   // (8, 32)
    dim3 bdim(256);

    repack_wconv<<<dim3((Cn * Cn * Tn + 255) / 256), bdim, 0, stream>>>(Wconv, wtb);

    gemm_wmma<false><<<gdim, bdim, 0, stream>>>(Q, nullptr, Wq, bq, qb, Mn, Cn, Cn);
    gemm_wmma<false><<<gdim, bdim, 0, stream>>>(Q, nullptr, Wk, bk, kb, Mn, Cn, Cn);
    gemm_wmma<false><<<gdim, bdim, 0, stream>>>(Q, nullptr, Wv, bv, vb, Mn, Cn, Cn);

    gemm_wmma<true><<<gdim, bdim, 0, stream>>>(vb, adj, wtb, bconv, convb, Mn, Cn, KCONV);
    gemm_wmma<false><<<gdim, bdim, 0, stream>>>(convb, nullptr, W2, b2, xlb, Mn, Cn, Cn);

    focus_kernel<<<dim3(Mn), bdim, 0, stream>>>(qb, scale_p);
    focus_kernel<<<dim3(Mn), bdim, 0, stream>>>(kb, scale_p);

    kmean_kernel<<<dim3((Bn * Cn) / 256), bdim, 0, stream>>>(kb, kmb);
    kv_kernel<<<dim3(Bn * Hn), bdim, 0, stream>>>(kb, vb, kvb);

    attn_kernel<<<dim3(Mn), bdim, 0, stream>>>(qb, kvb, kmb, xlb, convb);

    // losses: vh (raw v heads), kv, x4 (reshape of x)
    disag_kernel<<<dim3(32), dim3(32), 0, stream>>>(vb,   (long long)Nn * Cn, 64LL,               Cn, Nn, partb + 0);
    disag_kernel<<<dim3(32), dim3(32), 0, stream>>>(kvb,  (long long)Hn * Dn * Dn, (long long)Dn * Dn, Dn, Dn, partb + 32);
    disag_kernel<<<dim3(32), dim3(32), 0, stream>>>(convb,(long long)Nn * Cn, (long long)Nn * Dn, Dn, Nn, partb + 64);

    ln_kernel<<<dim3(Mn), bdim, 0, stream>>>(convb, ln_g, ln_b, wtb);
    gemm_wmma<false><<<gdim, bdim, 0, stream>>>(wtb, nullptr, Wo, bo, out, Mn, Cn, Cn);

    loss_final<<<dim3(1), dim3(32), 0, stream>>>(partb, out + MC);
}